// DiffTopkNet_69587060130315
// MI455X (gfx1250) — compile-verified
//
#include <hip/hip_runtime.h>

// DiffTopk via bitonic soft-sort, restructured:
//   X = L45 * ... * L1 * I  (each L symmetric 2x2-block blend matrix)
//   output rows = rows 511-v of X  ==>  propagate e_{511-v} backward through
//   the layers (L_t is symmetric, so row extraction == column propagation).
// Per batch: U[512][16] coefficient matrix in LDS; each layer applies a
// block-diagonal P (8 pairs -> 16x16 dense block) via V_WMMA_F32_16X16X4_F32.

#define BATCH    256
#define SIZE     512
#define K_OUT    16
#define NLAYERS  45
#define STEEP    10.0f
#define INV_PI   0.31830988618379067f

typedef float v2f __attribute__((ext_vector_type(2)));
typedef float v8f __attribute__((ext_vector_type(8)));

__global__ __launch_bounds__(256)
void difftopk_kernel(const float* __restrict__ vecs, float* __restrict__ out) {
    extern __shared__ float smem[];
    float* xs     = smem;                    // [512] values
    float* U      = smem + SIZE;             // [512][16] coefficient vectors
    float* alphaS = U + SIZE * K_OUT;        // [45][256] blend factors

    const int b   = blockIdx.x;
    const int tid = threadIdx.x;

    __builtin_prefetch(vecs + b * SIZE, 0, 0);   // global_prefetch_b8

    // ---- load x, seed U with flipped-identity rows (out row v == X row 511-v)
    xs[tid]       = vecs[b * SIZE + tid];
    xs[tid + 256] = vecs[b * SIZE + tid + 256];
    for (int e = tid; e < SIZE * K_OUT; e += 256) {
        int pos = e >> 4, v = e & 15;
        U[e] = (pos == (SIZE - 1 - v)) ? 1.0f : 0.0f;
    }
    __syncthreads();

    // ---- Phase 1: forward value pass, record all alphas (one pair per thread)
    {
        int lay = 0;
        for (int s = 1; s <= 9; ++s) {
            const int k = 1 << s;
            for (int jj = s - 1; jj >= 0; --jj) {
                const int j = 1 << jj;
                const int p = tid;
                const int i   = ((p & ~(j - 1)) << 1) | (p & (j - 1)); // p-th index with bit j clear
                const int prt = i | j;
                int l, r;
                if ((i & k) == 0) { l = i;   r = prt; }
                else              { l = prt; r = i;   }
                const float a = xs[l], c = xs[r];
                const float alpha = atanf((c - a) * STEEP) * INV_PI + 0.5f;
                xs[l] = alpha * a + (1.0f - alpha) * c;   // soft-min -> left
                xs[r] = alpha * c + (1.0f - alpha) * a;   // soft-max -> right
                alphaS[lay * 256 + p] = alpha;
                ++lay;
                __syncthreads();
            }
        }
    }

    // ---- Phase 2: backward propagation of the 16 row-vectors through the
    // layers in reverse. 8 waves x 4 groups x (8 pairs -> dense 16x16 block).
    const int wv   = tid >> 5;
    const int ln   = tid & 31;
    const int half = ln >> 4;      // 0: lanes 0-15, 1: lanes 16-31
    const int nn   = ln & 15;

    for (int t = NLAYERS - 1; t >= 0; --t) {
        // recover (k, j) of layer t from triangular enumeration
        int s = 1, base = 0;
        while (t >= base + s) { base += s; ++s; }
        const int k = 1 << s;
        const int j = 1 << (s - 1 - (t - base));

#pragma unroll
        for (int gi = 0; gi < 4; ++gi) {
            const int p0 = (wv * 4 + gi) * 8;    // 8 pairs of this group
            float al[8];
            int posL[8], posR[8];
#pragma unroll
            for (int i = 0; i < 8; ++i) {
                const int p   = p0 + i;
                const int ii  = ((p & ~(j - 1)) << 1) | (p & (j - 1));
                const int prt = ii | j;
                if ((ii & k) == 0) { posL[i] = ii;  posR[i] = prt; }
                else               { posL[i] = prt; posR[i] = ii;  }
                al[i] = alphaS[t * 256 + p];
            }

            v8f acc = {};
#if __has_builtin(__builtin_amdgcn_wmma_f32_16x16x4_f32)
            // D = P(16x16, block-diag of [[a,1-a],[1-a,a]]) x Ugroup(16x16),
            // chained over K in 4 chunks of 4 with V_WMMA_F32_16X16X4_F32.
            // A 16x4 layout : v0 = K0|K2 across lane halves, v1 = K1|K3.
            // B 4x16 layout : lanes 0-15 K0(v0),K1(v1); lanes 16-31 K2,K3.
#pragma unroll
            for (int kc = 0; kc < 4; ++kc) {
                const int   pL = half ? posL[2 * kc + 1] : posL[2 * kc];
                const int   pR = half ? posR[2 * kc + 1] : posR[2 * kc];
                const float av = half ? al[2 * kc + 1]   : al[2 * kc];
                v2f bfrag;
                bfrag.x = U[pL * K_OUT + nn];   // K = 4kc + 2*half     (even col)
                bfrag.y = U[pR * K_OUT + nn];   // K = 4kc + 2*half + 1 (odd col)
                const bool hit = ((nn >> 1) == (2 * kc + half));
                const bool odd = (nn & 1);
                v2f afrag;
                afrag.x = hit ? (odd ? 1.0f - av : av) : 0.0f;
                afrag.y = hit ? (odd ? av : 1.0f - av) : 0.0f;
                acc = __builtin_amdgcn_wmma_f32_16x16x4_f32(
                          false, afrag, false, bfrag, (short)0, acc, false, false);
            }
#else
            // scalar fallback: same D fragment layout, computed directly
#pragma unroll
            for (int r2 = 0; r2 < 8; ++r2) {
                const int m  = r2 + 8 * half;       // D row
                const int pm = m >> 1;
                const float av = al[pm];
                const float ul = U[posL[pm] * K_OUT + nn];
                const float ur = U[posR[pm] * K_OUT + nn];
                acc[r2] = (m & 1) ? ((1.0f - av) * ul + av * ur)
                                  : (av * ul + (1.0f - av) * ur);
            }
#endif
            // scatter D back: lane half h holds rows h*8 .. h*8+7, col nn
#pragma unroll
            for (int r2 = 0; r2 < 8; ++r2) {
                const int pm  = (r2 >> 1) + 4 * half;
                const int pos = (r2 & 1) ? posR[pm] : posL[pm];
                U[pos * K_OUT + nn] = acc[r2];
            }
        }
        __syncthreads();
    }

    // ---- Phase 3: write out[b][v][pos] = U[pos][v]  (32 KB per block)
    for (int e = tid; e < K_OUT * SIZE; e += 256) {
        const int v = e >> 9, pos = e & (SIZE - 1);
        out[b * (K_OUT * SIZE) + e] = U[pos * K_OUT + v];
    }
}

extern "C" void kernel_launch(void* const* d_in, const int* in_sizes, int n_in,
                              void* d_out, int out_size, void* d_ws, size_t ws_size,
                              hipStream_t stream) {
    (void)in_sizes; (void)n_in; (void)d_ws; (void)ws_size; (void)out_size;
    const float* vecs = (const float*)d_in[0];
    float* out = (float*)d_out;
    const size_t smem_bytes =
        (size_t)(SIZE + SIZE * K_OUT + NLAYERS * 256) * sizeof(float); // ~81 KB
    difftopk_kernel<<<dim3(BATCH), dim3(256), smem_bytes, stream>>>(vecs, out);
}